// BEVDistillationMaskGenerator_88115549045399
// MI455X (gfx1250) — compile-verified
//
#include <hip/hip_runtime.h>
#include <stdint.h>

typedef uint32_t u32;
typedef uint64_t u64;
typedef __attribute__((ext_vector_type(4))) u32 v4u;
typedef __attribute__((ext_vector_type(8))) int v8i;
typedef __attribute__((ext_vector_type(4))) int v4i;

#define FEAT 128
#define NBOX 2560

// LDS stage layout (float indices)
#define RR_OFF 0        // 512 rows x 5 floats (x,y,z,w,l)
#define RS_OFF 2560     // 512 floats
#define MG_OFF 3072     // 1024 x 8 floats (raw)
#define MS_OFF 11264    // 1024 floats
#define NU_OFF 12288    // 512 x 5 floats
#define MU_OFF 14848    // 512 x 5 floats
#define STAGE_F 17408   // 69632 bytes

#if defined(__has_builtin)
#if __has_builtin(__builtin_amdgcn_tensor_load_to_lds)
#define HAVE_TDM 1
#endif
#endif
#ifndef HAVE_TDM
#define HAVE_TDM 0
#endif

// Issue one TDM 2-D tile load: rows x tile_w elements (4B each), row pitch
// `stride` elements in global memory, compacted contiguously into LDS.
__device__ __forceinline__ void tdm_load_2d(u32 lds_byte, const void* gp,
                                            u32 tile_w, u32 rows, u32 stride) {
#if HAVE_TDM
  u64 ga = (u64)(uintptr_t)gp;
  v4u g0;
  g0.x = 1u;                              // count=1 (valid), user mode
  g0.y = lds_byte;                        // D#.lds_addr
  g0.z = (u32)ga;                         // global_addr[31:0]
  g0.w = (u32)(ga >> 32) | (2u << 30);    // global_addr[56:32] | type=2
  u32 w0 = (2u << 16);                                    // mask=0, data_size=4B
  u32 w1 = (stride & 0xFFFFu) << 16;                      // tensor_dim0[15:0]
  u32 w2 = (stride >> 16) | ((rows & 0xFFFFu) << 16);     // dim0[31:16] | tensor_dim1[15:0]
  u32 w3 = (rows >> 16) | (tile_w << 16);                 // dim1[31:16] | tile_dim0
  u32 w4 = rows & 0xFFFFu;                                // tile_dim1 (tile_dim2=0)
  u32 w5 = stride;                                        // tensor_dim0_stride[31:0]
  v8i g1;
  g1[0] = (int)w0; g1[1] = (int)w1; g1[2] = (int)w2; g1[3] = (int)w3;
  g1[4] = (int)w4; g1[5] = (int)w5; g1[6] = 0; g1[7] = 0;
  v4i z4 = {0, 0, 0, 0};
  v8i z8 = {0, 0, 0, 0, 0, 0, 0, 0};
  __builtin_amdgcn_tensor_load_to_lds(g0, g1, z4, z4, z8, 0);
#else
  (void)lds_byte; (void)gp; (void)tile_w; (void)rows; (void)stride;
#endif
}

// Native LDS float max atomic (no return): ds_max_num_f32 addr, data
__device__ __forceinline__ void lds_max_f32(float* p, float v) {
  u32 a = (u32)(uintptr_t)p;   // LDS aperture: low 32 bits == DS byte offset
  asm volatile("ds_max_num_f32 %0, %1" : : "v"(a), "v"(v) : "memory");
}

__global__ __launch_bounds__(1024)
void bev_mask_kernel(const float* __restrict__ rr, const float* __restrict__ rs,
                     const float* __restrict__ mg, const float* __restrict__ ms,
                     const float* __restrict__ nu, const float* __restrict__ mu,
                     float* __restrict__ out) {
  __shared__ float hm[FEAT * FEAT];   // 64 KB frame heatmap
  __shared__ float stage[STAGE_F];    // 68 KB staged box data

  const int f = blockIdx.x;
  const int tid = threadIdx.x;

  // Wave 0 issues the async tensor DMAs first (overlaps with heatmap clear).
  // TDM ignores EXEC; branch is wave-uniform.
  if (tid < 32) {
    u32 base = (u32)(uintptr_t)&stage[0];
    tdm_load_2d(base + RR_OFF * 4, rr + (size_t)f * 512 * 7, 5,    512, 7);
    tdm_load_2d(base + RS_OFF * 4, rs + (size_t)f * 512,     512,  1,   512);
    tdm_load_2d(base + MG_OFF * 4, mg + (size_t)f * 1024 * 8, 8192, 1,  8192);
    tdm_load_2d(base + MS_OFF * 4, ms + (size_t)f * 1024,    1024, 1,   1024);
    tdm_load_2d(base + NU_OFF * 4, nu + (size_t)f * 512 * 8, 5,    512, 8);
    tdm_load_2d(base + MU_OFF * 4, mu + (size_t)f * 512 * 8, 5,    512, 8);
  }

  for (int i = tid; i < FEAT * FEAT; i += blockDim.x) hm[i] = 0.0f;

#if HAVE_TDM
  if (tid < 32) __builtin_amdgcn_s_wait_tensorcnt(0);
#else
  // Fallback: cooperative copy (kept functionally equivalent)
  for (int i = tid; i < 512; i += blockDim.x) {
    for (int k = 0; k < 5; ++k) {
      stage[RR_OFF + i * 5 + k] = rr[(size_t)f * 512 * 7 + i * 7 + k];
      stage[NU_OFF + i * 5 + k] = nu[(size_t)f * 512 * 8 + i * 8 + k];
      stage[MU_OFF + i * 5 + k] = mu[(size_t)f * 512 * 8 + i * 8 + k];
    }
    stage[RS_OFF + i] = rs[(size_t)f * 512 + i];
  }
  for (int i = tid; i < 1024 * 8; i += blockDim.x)
    stage[MG_OFF + i] = mg[(size_t)f * 1024 * 8 + i];
  for (int i = tid; i < 1024; i += blockDim.x)
    stage[MS_OFF + i] = ms[(size_t)f * 1024 + i];
#endif
  __syncthreads();

  const int wave = tid >> 5;
  const int lane = tid & 31;
  const int nw = (int)(blockDim.x >> 5);
  const float INV_SCALE = 1.25f;  // 1 / (0.1 * 8)

  for (int b = wave; b < NBOX; b += nw) {
    float x, y, wbox, lbox, val;
    if (b < 512) {
      const float* p = &stage[RR_OFF + b * 5];
      x = p[0]; y = p[1]; wbox = p[3]; lbox = p[4];
      val = stage[RS_OFF + b];
    } else if (b < 1536) {
      int i = b - 512;
      const float* p = &stage[MG_OFF + i * 8];
      x = p[0]; y = p[1]; wbox = p[3]; lbox = p[4];
      int cls = (int)p[7];
      bool small = (cls == 5) | (cls == 6) | (cls == 8) | (cls == 9);
      val = small ? (stage[MS_OFF + i] + 1.0f) * 0.5f : 0.5f;
    } else if (b < 2048) {
      int i = b - 1536;
      const float* p = &stage[NU_OFF + i * 5];
      x = p[0]; y = p[1]; wbox = p[3]; lbox = p[4]; val = 0.4f;
    } else {
      int i = b - 2048;
      const float* p = &stage[MU_OFF + i * 5];
      x = p[0]; y = p[1]; wbox = p[3]; lbox = p[4]; val = 0.2f;
    }

    float wf = wbox * INV_SCALE, lf = lbox * INV_SCALE;
    bool valid = (wf > 0.0f) & (lf > 0.0f) & (wf <= 1000.0f) & (lf <= 1000.0f);

    // gaussian_radius(height=lf, width=wf, overlap=0.1)
    float h = lf, wd = wf;
    float b1 = h + wd, c1 = wd * h * (0.9f / 1.1f);
    float r1 = 0.5f * (b1 + sqrtf(fmaxf(b1 * b1 - 4.0f * c1, 0.0f)));
    float b2 = 2.0f * (h + wd), c2 = 0.9f * wd * h;
    float r2 = 0.5f * (b2 + sqrtf(fmaxf(b2 * b2 - 16.0f * c2, 0.0f)));
    float b3 = -0.2f * (h + wd), c3 = -0.9f * wd * h;
    float r3 = 0.5f * (b3 + sqrtf(fmaxf(b3 * b3 - 1.6f * c3, 0.0f)));
    float rfl = fminf(r1, fminf(r2, r3));
    int r = (int)rfl;              // trunc, matches astype(int32) for r>=0
    if (r < 2) r = 2;
    if (r > 32) r = 32;

    int cx = (int)((x + 51.2f) * INV_SCALE);
    int cy = (int)((y + 51.2f) * INV_SCALE);
    valid = valid & (cx >= 0) & (cx < FEAT) & (cy >= 0) & (cy < FEAT);
    if (!valid) continue;          // uniform per wave

    int D = 2 * r + 1;
    int npix = D * D;
    float inv2s2 = 18.0f / (float)(D * D);   // 1/(2*sigma^2), sigma=D/6
    for (int p = lane; p < npix; p += 32) {
      int dy = p / D - r;
      int dx = p - (dy + r) * D - r;
      int yy = cy + dy, xx = cx + dx;
      if ((unsigned)yy < (unsigned)FEAT && (unsigned)xx < (unsigned)FEAT) {
        float g = val * __expf(-(float)(dx * dx + dy * dy) * inv2s2);
        lds_max_f32(&hm[yy * FEAT + xx], g);   // ds_max_num_f32
      }
    }
  }

  // Inline-asm DS atomics are untracked by the compiler: drain DScnt before
  // signaling the barrier so all waves' maxes are visible afterwards.
  asm volatile("s_wait_dscnt 0x0" : : : "memory");
  __syncthreads();

  float* o = out + (size_t)f * FEAT * FEAT;
  for (int i = tid; i < FEAT * FEAT; i += blockDim.x) o[i] = hm[i];
}

extern "C" void kernel_launch(void* const* d_in, const int* in_sizes, int n_in,
                              void* d_out, int out_size, void* d_ws, size_t ws_size,
                              hipStream_t stream) {
  (void)in_sizes; (void)n_in; (void)out_size; (void)d_ws; (void)ws_size;
  const float* rr = (const float*)d_in[0];  // refined_rois    (4,512,7)
  const float* rs = (const float*)d_in[1];  // refined_scores  (4,512)
  const float* mg = (const float*)d_in[2];  // medium_gts      (4,1024,8)
  const float* ms = (const float*)d_in[3];  // medium_scores   (4,1024)
  const float* nu = (const float*)d_in[4];  // near_unmatched  (4,512,8)
  const float* mu = (const float*)d_in[5];  // medium_unmatched(4,512,8)
  float* out = (float*)d_out;               // (4,1,128,128)
  bev_mask_kernel<<<4, 1024, 0, stream>>>(rr, rs, mg, ms, nu, mu, out);
}